// CapsNet_72945724555674
// MI455X (gfx1250) — compile-verified
//
#include <hip/hip_runtime.h>
#include <hip/hip_bf16.h>
#include <math.h>
#include <stdint.h>

typedef __attribute__((ext_vector_type(2))) float v2f;
typedef __attribute__((ext_vector_type(8))) float v8f;
typedef unsigned int u32x4 __attribute__((ext_vector_type(4)));
typedef int i32x4 __attribute__((ext_vector_type(4)));
typedef int i32x8 __attribute__((ext_vector_type(8)));

#define B_SZ   32
#define CONV_OC 256
#define IMG    28
#define OUTW   20
#define KW     9
#define FLAT_K 102400        // 256*20*20
#define FC_N   9216
#define NCAPS  1152
#define CAPD   8
#define NDIGIT 10
#define DDIM   16
#define KT     256           // TDM tile K-width (floats)
#define PITCH  (KT + 4)      // LDS row pitch after TDM padding (4 dwords / row)

// ---------------------------------------------------------------------------
// 1) conv 9x9 valid, 1->256, + bias + relu.  One block per (b, oc).
// ---------------------------------------------------------------------------
__global__ void __launch_bounds__(256)
conv_relu_kernel(const float* __restrict__ x, const float* __restrict__ w,
                 const float* __restrict__ bias, float* __restrict__ flat) {
  __shared__ float img[IMG * IMG];
  __shared__ float wt[KW * KW];
  const int b  = blockIdx.x / CONV_OC;
  const int oc = blockIdx.x % CONV_OC;
  const int tid = threadIdx.x;
  for (int i = tid; i < IMG * IMG; i += 256) img[i] = x[b * IMG * IMG + i];
  if (tid < KW * KW) wt[tid] = w[oc * KW * KW + tid];
  __syncthreads();
  const float bv = bias[oc];
  for (int o = tid; o < OUTW * OUTW; o += 256) {
    const int oy = o / OUTW, ox = o % OUTW;
    float acc = bv;
    #pragma unroll
    for (int ky = 0; ky < KW; ++ky)
      #pragma unroll
      for (int kx = 0; kx < KW; ++kx)
        acc = fmaf(img[(oy + ky) * IMG + ox + kx], wt[ky * KW + kx], acc);
    flat[((size_t)b * CONV_OC + oc) * (OUTW * OUTW) + o] = fmaxf(acc, 0.f);
  }
}

// ---------------------------------------------------------------------------
// TDM: DMA a 16-row x KT-col f32 tile of fc_w into LDS with 4-dword row pad.
// D# built per CDNA5 ISA 8.3/8.4:
//  group0: count=1 | lds_addr | global_addr(57b) | type=2
//  group1: data_size=2(4B), pad_enable, pad_interval=7 (256 dw), pad_amount=3
//          (4 dw), tensor_dim0=102400, tensor_dim1=9216, tile 256x16,
//          tensor_dim0_stride=102400
// ---------------------------------------------------------------------------
__device__ __forceinline__ void tdm_load_tile(uint32_t lds_addr, uint64_t gaddr) {
  u32x4 g0;
  g0.x = 0x1u;                                             // count=1, user mode
  g0.y = lds_addr;                                         // LDS byte address
  g0.z = (uint32_t)(gaddr & 0xFFFFFFFFu);                  // global_addr[31:0]
  g0.w = ((uint32_t)(gaddr >> 32) & 0x01FFFFFFu) | 0x80000000u; // [56:32] | type=2
  i32x8 g1;
  g1[0] = (int)(0x00020000u        // data_size=2 (4B)
              | (1u << 20)         // pad_enable
              | (7u << 22)         // pad_interval: 256 dwords (one row)
              | (3u << 25));       // pad_amount: 4 dwords
  g1[1] = (int)((FLAT_K & 0xFFFFu) << 16);                          // dim0 lo16 @48
  g1[2] = (int)(((FLAT_K >> 16) & 0xFFFFu) | ((FC_N & 0xFFFFu) << 16)); // dim0 hi | dim1 lo
  g1[3] = (int)(((FC_N >> 16) & 0xFFFFu) | ((uint32_t)KT << 16));   // dim1 hi | tile_dim0
  g1[4] = 16;                                                       // tile_dim1 (tile_dim2=0)
  g1[5] = FLAT_K;                                                   // dim0_stride lo32
  g1[6] = 0;                                                        // stride hi / dim1_stride
  g1[7] = 0;
  i32x4 z4 = {0, 0, 0, 0};
#if __clang_major__ >= 23
  i32x8 z8 = {0, 0, 0, 0, 0, 0, 0, 0};
  __builtin_amdgcn_tensor_load_to_lds(g0, g1, z4, z4, z8, 0);
#else
  __builtin_amdgcn_tensor_load_to_lds(g0, g1, z4, z4, 0);
#endif
}

// ---------------------------------------------------------------------------
// 2) FC 102400 -> 9216.  pc[b,n] = flat[b,:].fc_w[n,:] + fc_b[n]
//    576 blocks own a 16-column N-tile; TDM double-buffers 16xKT weight tiles
//    into LDS (padded pitch -> conflict-free ds_load_b64); 8 waves split each
//    tile's K; V_WMMA_F32_16X16X4_F32 with two 16x16 accumulators (M=32);
//    cross-wave reduce in LDS.
// ---------------------------------------------------------------------------
__global__ void __launch_bounds__(256)
fc_wmma_kernel(const float* __restrict__ flat, const float* __restrict__ fcw,
               const float* __restrict__ fcb, float* __restrict__ pc) {
  __shared__ float btile[2][16 * PITCH];       // 2 x 16.25 KB
  __shared__ float red[8][2][8][32];           // 16 KB
  const int nbase = blockIdx.x * 16;
  const int tid  = threadIdx.x;
  const int wave = tid >> 5;
  const int lane = tid & 31;
  const int half = lane >> 4;                  // 0: K+0/K+1,  1: K+2/K+3
  const int l16  = lane & 15;
  const int kOff = half * 2;

  const float* arow0 = flat + (size_t)l16 * FLAT_K;          // A rows 0..15
  const float* arow1 = flat + (size_t)(l16 + 16) * FLAT_K;   // A rows 16..31

  const uint64_t gbase = (uint64_t)(uintptr_t)(fcw + (size_t)nbase * FLAT_K);
  const uint32_t lds0 = (uint32_t)(uintptr_t)&btile[0][0];
  const uint32_t lds1 = (uint32_t)(uintptr_t)&btile[1][0];

  if (wave == 0) tdm_load_tile(lds0, gbase);

  v8f c0 = {}; v8f c1 = {};
  const int T = FLAT_K / KT;                   // 400 tiles
  for (int t = 0; t < T; ++t) {
    if (wave == 0) {
      if (t + 1 < T) {
        tdm_load_tile(((t + 1) & 1) ? lds1 : lds0,
                      gbase + (uint64_t)(t + 1) * (KT * 4));
        __builtin_amdgcn_s_wait_tensorcnt(1);  // tile t complete
      } else {
        __builtin_amdgcn_s_wait_tensorcnt(0);
      }
    }
    __syncthreads();                           // tile t visible to all waves
    const float* bl = &btile[t & 1][0] + l16 * PITCH + wave * 32 + kOff;
    const int kk0 = t * KT + wave * 32;
    #pragma unroll
    for (int s = 0; s < 8; ++s) {
      const int c = s * 4;
      v2f a0 = *(const v2f*)(arow0 + kk0 + c + kOff);
      v2f a1 = *(const v2f*)(arow1 + kk0 + c + kOff);
      v2f bb = *(const v2f*)(bl + c);
      c0 = __builtin_amdgcn_wmma_f32_16x16x4_f32(false, a0, false, bb, (short)0, c0, false, false);
      c1 = __builtin_amdgcn_wmma_f32_16x16x4_f32(false, a1, false, bb, (short)0, c1, false, false);
    }
    __syncthreads();                           // all waves done reading tile t
  }
  #pragma unroll
  for (int v = 0; v < 8; ++v) {
    red[wave][0][v][lane] = c0[v];
    red[wave][1][v][lane] = c1[v];
  }
  __syncthreads();
  // 512 output elements; 2 per thread; reduce across 8 waves, add bias.
  for (int e = tid; e < 512; e += 256) {
    const int tile = e >> 8;
    const int rem  = e & 255;
    const int v    = rem >> 5;
    const int ln   = rem & 31;
    float s = 0.f;
    #pragma unroll
    for (int wv = 0; wv < 8; ++wv) s += red[wv][tile][v][ln];
    const int m = v + 8 * (ln >> 4) + 16 * tile;      // C/D layout: M=v(+8), N=lane%16
    const int n = nbase + (ln & 15);
    pc[(size_t)m * FC_N + n] = s + fcb[n];
  }
}

// ---------------------------------------------------------------------------
// 3) squash primary caps: [32,1152,8] along last dim
// ---------------------------------------------------------------------------
__global__ void __launch_bounds__(256)
squash_primary_kernel(const float* __restrict__ pc, float* __restrict__ caps) {
  const int i = blockIdx.x * 256 + threadIdx.x;    // capsule id, exactly 36864
  const float* p = pc + (size_t)i * 8;
  float4 a = *(const float4*)p;
  float4 b = *(const float4*)(p + 4);
  float n2 = a.x*a.x + a.y*a.y + a.z*a.z + a.w*a.w
           + b.x*b.x + b.y*b.y + b.z*b.z + b.w*b.w;
  float scale = n2 / (1.f + n2);
  float inv = 1.f / sqrtf(n2);
  float4 oa, ob;
  oa.x = scale * (a.x * inv + 1e-8f); oa.y = scale * (a.y * inv + 1e-8f);
  oa.z = scale * (a.z * inv + 1e-8f); oa.w = scale * (a.w * inv + 1e-8f);
  ob.x = scale * (b.x * inv + 1e-8f); ob.y = scale * (b.y * inv + 1e-8f);
  ob.z = scale * (b.z * inv + 1e-8f); ob.w = scale * (b.w * inv + 1e-8f);
  *(float4*)(caps + (size_t)i * 8) = oa;
  *(float4*)(caps + (size_t)i * 8 + 4) = ob;
}

// ---------------------------------------------------------------------------
// 4) u_hat[b,n,p,d] = sum_i W[n,p,d,i]*caps[b,p,i] via WMMA f32 16x16x4.
//    One wave per (n,p); K=8 -> two WMMA steps; M=32 -> two tiles.
//    11520 pairs = 1440 blocks * 8 waves exactly (EXEC always all-ones).
// ---------------------------------------------------------------------------
__global__ void __launch_bounds__(256)
uhat_wmma_kernel(const float* __restrict__ caps,   // [32,1152,8]
                 const float* __restrict__ W,      // [10,1152,16,8]
                 float* __restrict__ uhat) {       // [32,10,1152,16]
  const int pair = blockIdx.x * 8 + (threadIdx.x >> 5);
  const int n = pair / NCAPS;
  const int p = pair % NCAPS;
  const int lane = threadIdx.x & 31;
  const int half = lane >> 4;
  const int l16  = lane & 15;
  const float* a0p = caps + ((size_t)l16 * NCAPS + p) * CAPD;         // rows b=0..15
  const float* a1p = caps + ((size_t)(l16 + 16) * NCAPS + p) * CAPD;  // rows b=16..31
  const float* bp  = W + (((size_t)n * NCAPS + p) * DDIM + l16) * CAPD;
  v8f c0 = {}; v8f c1 = {};
  #pragma unroll
  for (int step = 0; step < 2; ++step) {
    const int i = step * 4 + half * 2;
    v2f a0 = *(const v2f*)(a0p + i);
    v2f a1 = *(const v2f*)(a1p + i);
    v2f bb = *(const v2f*)(bp + i);
    c0 = __builtin_amdgcn_wmma_f32_16x16x4_f32(false, a0, false, bb, (short)0, c0, false, false);
    c1 = __builtin_amdgcn_wmma_f32_16x16x4_f32(false, a1, false, bb, (short)0, c1, false, false);
  }
  const size_t base = ((size_t)n * NCAPS + p) * DDIM + l16;   // d = lane%16
  #pragma unroll
  for (int v = 0; v < 8; ++v) {
    const int b0 = v + 8 * half;
    uhat[(size_t)b0 * (NDIGIT * NCAPS * DDIM) + base] = c0[v];
    uhat[(size_t)(b0 + 16) * (NDIGIT * NCAPS * DDIM) + base] = c1[v];
  }
}

// ---------------------------------------------------------------------------
// 5) dynamic routing (3 iters) + logits + argmax mask, one block per batch elem
// ---------------------------------------------------------------------------
__global__ void __launch_bounds__(256)
routing_kernel(const float* __restrict__ uhat,   // [32,10,1152,16]
               float* __restrict__ out_logits,   // [32,10]
               float* __restrict__ out_digit,    // [32,10,16]
               float* __restrict__ out_c,        // [32,10,1152]
               float* __restrict__ out_b,        // [32,10,1152]
               float* __restrict__ masked) {     // [32,160]
  __shared__ float bl[NDIGIT * NCAPS];           // 46 KB
  __shared__ float sv[NDIGIT * DDIM];
  __shared__ float vv[NDIGIT * DDIM];
  __shared__ float nrm[NDIGIT];
  __shared__ float lgs[NDIGIT];
  __shared__ int amax;
  const int b = blockIdx.x;
  const int tid = threadIdx.x;
  const float* ub = uhat + (size_t)b * NDIGIT * NCAPS * DDIM;
  float* cb = out_c + (size_t)b * NDIGIT * NCAPS;   // scratch during iters; final c remains

  for (int i = tid; i < NDIGIT * NCAPS; i += 256) bl[i] = 0.f;
  __syncthreads();

  for (int iter = 0; iter < 3; ++iter) {
    // c = softmax(b, axis=n) per column p
    for (int p = tid; p < NCAPS; p += 256) {
      float mx = bl[p];
      #pragma unroll
      for (int n = 1; n < NDIGIT; ++n) mx = fmaxf(mx, bl[n * NCAPS + p]);
      float e[NDIGIT]; float ssum = 0.f;
      #pragma unroll
      for (int n = 0; n < NDIGIT; ++n) { e[n] = __expf(bl[n * NCAPS + p] - mx); ssum += e[n]; }
      const float r = 1.f / ssum;
      #pragma unroll
      for (int n = 0; n < NDIGIT; ++n) cb[n * NCAPS + p] = e[n] * r;
    }
    __threadfence_block();
    __syncthreads();
    // s[n,d] = sum_p c[n,p] * u_hat[b,n,p,d]   (160 threads; d contiguous -> coalesced)
    if (tid < NDIGIT * DDIM) {
      const int n = tid >> 4, d = tid & 15;
      const float* up = ub + (size_t)n * NCAPS * DDIM + d;
      const float* cp = cb + n * NCAPS;
      float acc = 0.f;
      for (int p = 0; p < NCAPS; ++p) acc = fmaf(cp[p], up[(size_t)p * DDIM], acc);
      sv[tid] = acc;
    }
    __syncthreads();
    if (tid < NDIGIT) {
      float n2 = 0.f;
      #pragma unroll
      for (int d = 0; d < DDIM; ++d) { const float v = sv[tid * DDIM + d]; n2 = fmaf(v, v, n2); }
      nrm[tid] = n2;
    }
    __syncthreads();
    if (tid < NDIGIT * DDIM) {
      const float n2 = nrm[tid >> 4];
      const float scale = n2 / (1.f + n2);
      vv[tid] = scale * (sv[tid] / sqrtf(n2) + 1e-8f);
    }
    __syncthreads();
    if (iter < 2) {
      // b += sum_d u_hat * v
      for (int idx = tid; idx < NDIGIT * NCAPS; idx += 256) {
        const int n = idx / NCAPS, p = idx % NCAPS;
        const float* up = ub + ((size_t)n * NCAPS + p) * DDIM;
        const float* vp = vv + n * DDIM;
        float dot = 0.f;
        #pragma unroll
        for (int d = 0; d < DDIM; ++d) dot = fmaf(up[d], vp[d], dot);
        bl[idx] += dot;
      }
      __syncthreads();
    }
  }
  // outputs
  if (tid < NDIGIT * DDIM) out_digit[(size_t)b * 160 + tid] = vv[tid];
  if (tid < NDIGIT) {
    float n2 = 0.f;
    #pragma unroll
    for (int d = 0; d < DDIM; ++d) { const float v = vv[tid * DDIM + d]; n2 = fmaf(v, v, n2); }
    const float l = sqrtf(n2);
    lgs[tid] = l;
    out_logits[b * NDIGIT + tid] = l;
  }
  __syncthreads();
  if (tid == 0) {
    int bi = 0; float bv = lgs[0];
    #pragma unroll
    for (int n = 1; n < NDIGIT; ++n) if (lgs[n] > bv) { bv = lgs[n]; bi = n; }
    amax = bi;
  }
  __syncthreads();
  if (tid < NDIGIT * DDIM)
    masked[(size_t)b * 160 + tid] = ((tid >> 4) == amax) ? vv[tid] : 0.f;
  for (int i = tid; i < NDIGIT * NCAPS; i += 256)
    out_b[(size_t)b * NDIGIT * NCAPS + i] = bl[i];
  // out_c already holds final-iteration softmax
}

// ---------------------------------------------------------------------------
// 6) decoder FC layers (tiny): act 0=relu, 1=sigmoid
// ---------------------------------------------------------------------------
__global__ void __launch_bounds__(256)
fc_act_kernel(const float* __restrict__ in, const float* __restrict__ w,
              const float* __restrict__ bias, float* __restrict__ out,
              int K, int N, int act) {
  const int i = blockIdx.x * 256 + threadIdx.x;
  if (i >= B_SZ * N) return;
  const int b = i / N, n = i % N;
  const float4* ip = (const float4*)(in + (size_t)b * K);
  const float4* wp = (const float4*)(w + (size_t)n * K);
  float acc = bias[n];
  const int K4 = K >> 2;
  for (int k = 0; k < K4; ++k) {
    const float4 a = ip[k], ww = wp[k];
    acc = fmaf(a.x, ww.x, acc); acc = fmaf(a.y, ww.y, acc);
    acc = fmaf(a.z, ww.z, acc); acc = fmaf(a.w, ww.w, acc);
  }
  out[i] = (act == 0) ? fmaxf(acc, 0.f) : (1.f / (1.f + __expf(-acc)));
}

// ---------------------------------------------------------------------------
extern "C" void kernel_launch(void* const* d_in, const int* in_sizes, int n_in,
                              void* d_out, int out_size, void* d_ws, size_t ws_size,
                              hipStream_t stream) {
  const float* x      = (const float*)d_in[0];
  const float* conv_w = (const float*)d_in[1];
  const float* conv_b = (const float*)d_in[2];
  const float* fc_w   = (const float*)d_in[3];
  const float* fc_b   = (const float*)d_in[4];
  const float* W      = (const float*)d_in[5];   // [1,10,1152,16,8] -> use W[0]
  const float* d1_w   = (const float*)d_in[6];
  const float* d1_b   = (const float*)d_in[7];
  const float* d2_w   = (const float*)d_in[8];
  const float* d2_b   = (const float*)d_in[9];
  const float* d3_w   = (const float*)d_in[10];
  const float* d3_b   = (const float*)d_in[11];

  float* out = (float*)d_out;
  float* o_logits = out;                         // 32*10      = 320
  float* o_rec    = out + 320;                   // 32*784     = 25088
  float* o_prim   = out + 25408;                 // 32*1152*8  = 294912
  float* o_digit  = out + 320320;                // 32*10*16   = 5120
  float* o_c      = out + 325440;                // 32*10*1152 = 368640
  float* o_b      = out + 694080;                // 32*10*1152 = 368640

  float* ws     = (float*)d_ws;                  // ~38.1 MB total
  float* flat   = ws;                            // 32*102400  = 3276800
  float* pc     = flat + 3276800;                // 32*9216    = 294912
  float* uhat   = pc + 294912;                   // 32*10*1152*16 = 5898240
  float* masked = uhat + 5898240;                // 32*160     = 5120
  float* h1     = masked + 5120;                 // 32*512     = 16384
  float* h2     = h1 + 16384;                    // 32*1024    = 32768

  conv_relu_kernel<<<B_SZ * CONV_OC, 256, 0, stream>>>(x, conv_w, conv_b, flat);
  fc_wmma_kernel<<<FC_N / 16, 256, 0, stream>>>(flat, fc_w, fc_b, pc);
  squash_primary_kernel<<<(B_SZ * NCAPS) / 256, 256, 0, stream>>>(pc, o_prim);
  uhat_wmma_kernel<<<(NDIGIT * NCAPS) / 8, 256, 0, stream>>>(o_prim, W, uhat);
  routing_kernel<<<B_SZ, 256, 0, stream>>>(uhat, o_logits, o_digit, o_c, o_b, masked);
  fc_act_kernel<<<(B_SZ * 512 + 255) / 256, 256, 0, stream>>>(masked, d1_w, d1_b, h1, 160, 512, 0);
  fc_act_kernel<<<(B_SZ * 1024 + 255) / 256, 256, 0, stream>>>(h1, d2_w, d2_b, h2, 512, 1024, 0);
  fc_act_kernel<<<(B_SZ * 784 + 255) / 256, 256, 0, stream>>>(h2, d3_w, d3_b, o_rec, 1024, 784, 1);
}